// PatchLevelPruner_8641474200510
// MI455X (gfx1250) — compile-verified
//
#include <hip/hip_runtime.h>
#include <math.h>

typedef __attribute__((ext_vector_type(2))) float v2f;
typedef __attribute__((ext_vector_type(8))) float v8f;

#define NB    16      // batch
#define HIMG  128
#define WIMG  128
#define CDIM  256
#define HID   64
#define PH    8
#define NHP   16      // patches per spatial dim
#define NP    256     // patches per image
#define NK    179     // kept patches
#define NKH   12
#define NKW   16
#define WNEW  (NKW*PH)        // 128
#define NROW  (NKH*PH*WNEW)   // 12288 output tokens per image
#define XPAD  260             // padded LDS row stride (words): 260 % 64 == 4 -> conflict-free A frags

// ---------------------------------------------------------------------------
// Kernel 1: per-patch scorer.  One workgroup = one (batch, patch).
// H = relu(X @ W1 + b1) via V_WMMA_F32_16X16X4_F32, then fused dot with W2,
// sigmoid, mean over the 64 patch tokens.
//
// W1 is staged pair-interleaved: W1p[((k>>1)*HID + n)*2 + (k&1)], so a B
// fragment (rows ka, ka+1 / col n) is ONE aligned 8-byte LDS load that lands
// directly in the even VGPR pair WMMA wants (no v_mov repacking), and the 4
// N-tiles fuse into two ds_load_2addr_b64.
// ---------------------------------------------------------------------------
__global__ __launch_bounds__(128)
void score_kernel(const float* __restrict__ tokens,
                  const float* __restrict__ W1,
                  const float* __restrict__ b1,
                  const float* __restrict__ W2,
                  const float* __restrict__ b2,
                  float* __restrict__ scores)
{
    __shared__ float Xs[64 * XPAD];     // 66,560 B : 64 tokens x 256 ch (padded)
    __shared__ float W1p[CDIM * HID];   // 65,536 B : pair-interleaved 256 x 64
    __shared__ float Hred[64 * 16];     // cross-lane reduction buffer
    __shared__ float TokImp[64];

    const int wg  = blockIdx.x;     // b*NP + p
    const int b   = wg >> 8;
    const int p   = wg & 255;
    const int py  = p >> 4;
    const int px  = p & 15;
    const int tid = threadIdx.x;

    // ---- stage X tile (coalesced: 64 consecutive float4 per token row) ----
    const size_t tokBase = (size_t)b * (HIMG * WIMG) * CDIM;
    for (int idx = tid; idx < 64 * (CDIM / 4); idx += 128) {
        const int tok = idx >> 6;           // 0..63
        const int c4  = idx & 63;
        const int i   = tok >> 3;
        const int j   = tok & 7;
        const int row = (py * PH + i) * WIMG + (px * PH + j);
        const float4 v = *(const float4*)(tokens + tokBase + (size_t)row * CDIM + c4 * 4);
        float* dst = &Xs[tok * XPAD + c4 * 4];
        dst[0] = v.x; dst[1] = v.y; dst[2] = v.z; dst[3] = v.w;
    }
    // ---- stage W1, pair-interleaved over K ----
    for (int idx = tid; idx < (CDIM * HID) / 4; idx += 128) {
        const float4 v = *(const float4*)(W1 + idx * 4);
        const int k  = (idx * 4) / HID;
        const int n  = (idx * 4) % HID;     // multiple of 4
        const int kp = k >> 1;
        const int kl = k & 1;
        float* dst = &W1p[((kp * HID) + n) * 2 + kl];
        dst[0] = v.x; dst[2] = v.y; dst[4] = v.z; dst[6] = v.w;
    }
    __syncthreads();

    const int wave = tid >> 5;          // 0..3 -> M tile
    const int lane = tid & 31;
    const int lr   = lane & 15;
    const int lh   = lane >> 4;         // lane half -> +2 in K (i.e. +1 in K-pairs)
    const int m0   = wave * 16;

    v8f acc0 = {}, acc1 = {}, acc2 = {}, acc3 = {};
    const float* xrow = &Xs[(m0 + lr) * XPAD];

    for (int k = 0; k < CDIM; k += 4) {
        const int ka = k + 2 * lh;                       // even
        const v2f a = *(const v2f*)&xrow[ka];            // A frag: 1x b64
        const float* wp = &W1p[(((k >> 1) + lh) * HID + lr) * 2];
        const v2f b0 = *(const v2f*)(wp + 0);            // 4 B frags: fuse to 2x 2addr_b64
        const v2f b1v = *(const v2f*)(wp + 32);
        const v2f b2v = *(const v2f*)(wp + 64);
        const v2f b3v = *(const v2f*)(wp + 96);
        acc0 = __builtin_amdgcn_wmma_f32_16x16x4_f32(false, a, false, b0,  (short)0, acc0, false, false);
        acc1 = __builtin_amdgcn_wmma_f32_16x16x4_f32(false, a, false, b1v, (short)0, acc1, false, false);
        acc2 = __builtin_amdgcn_wmma_f32_16x16x4_f32(false, a, false, b2v, (short)0, acc2, false, false);
        acc3 = __builtin_amdgcn_wmma_f32_16x16x4_f32(false, a, false, b3v, (short)0, acc3, false, false);
    }

    // ---- bias + relu + dot(W2), fused on accumulators ----
    float part[8];
#pragma unroll
    for (int v = 0; v < 8; ++v) part[v] = 0.0f;
    {
        const float bn = b1[0 + lr], wn = W2[0 + lr];
#pragma unroll
        for (int v = 0; v < 8; ++v) { float h = acc0[v] + bn; h = h > 0.f ? h : 0.f; part[v] += h * wn; }
    }
    {
        const float bn = b1[16 + lr], wn = W2[16 + lr];
#pragma unroll
        for (int v = 0; v < 8; ++v) { float h = acc1[v] + bn; h = h > 0.f ? h : 0.f; part[v] += h * wn; }
    }
    {
        const float bn = b1[32 + lr], wn = W2[32 + lr];
#pragma unroll
        for (int v = 0; v < 8; ++v) { float h = acc2[v] + bn; h = h > 0.f ? h : 0.f; part[v] += h * wn; }
    }
    {
        const float bn = b1[48 + lr], wn = W2[48 + lr];
#pragma unroll
        for (int v = 0; v < 8; ++v) { float h = acc3[v] + bn; h = h > 0.f ? h : 0.f; part[v] += h * wn; }
    }
    // C/D layout: row = m0 + v + 8*lh, this lane covered cols {lr, lr+16, lr+32, lr+48}
#pragma unroll
    for (int v = 0; v < 8; ++v)
        Hred[(m0 + v + 8 * lh) * 16 + lr] = part[v];
    __syncthreads();

    if (tid < 64) {
        float s = 0.0f;
#pragma unroll
        for (int l = 0; l < 16; ++l) s += Hred[tid * 16 + l];
        const float z = s + b2[0];
        TokImp[tid] = 1.0f / (1.0f + expf(-z));
    }
    __syncthreads();

    if (tid == 0) {
        float s = 0.0f;
        for (int t = 0; t < 64; ++t) s += TokImp[t];
        scores[wg] = s * (1.0f / 64.0f);
    }
}

// ---------------------------------------------------------------------------
// Kernel 2: rank-based top-k with JAX tie semantics, compacted in ascending
// index order (== jnp.sort(top_idx)).
// ---------------------------------------------------------------------------
__global__ __launch_bounds__(256)
void topk_kernel(const float* __restrict__ scores, int* __restrict__ top_idx)
{
    __shared__ float s[NP];
    __shared__ int keep[NP];
    const int b = blockIdx.x;
    const int i = threadIdx.x;
    s[i] = scores[b * NP + i];
    __syncthreads();
    const float si = s[i];
    int rank = 0;
    for (int j = 0; j < NP; ++j) {
        const float sj = s[j];
        rank += (sj > si) || (sj == si && j < i);
    }
    keep[i] = (rank < NK) ? 1 : 0;
    __syncthreads();
    if (keep[i]) {
        int slot = 0;
        for (int j = 0; j < i; ++j) slot += keep[j];
        top_idx[b * NK + slot] = i;
    }
}

// ---------------------------------------------------------------------------
// Kernel 3: gather kept patches into the padded nKh x nKw grid.
// One workgroup per (batch, output patch slot). float4, fully coalesced,
// with a software prefetch (global_prefetch_b8) one iteration ahead.
// ---------------------------------------------------------------------------
__global__ __launch_bounds__(256)
void gather_kernel(const float* __restrict__ tokens,
                   const int* __restrict__ top_idx,
                   float* __restrict__ out)
{
    const int wg = blockIdx.x;          // b*(NKH*NKW) + q
    const int b  = wg / (NKH * NKW);
    const int q  = wg % (NKH * NKW);
    const int kh = q >> 4;
    const int kw = q & 15;
    const size_t outBase = (size_t)b * NROW * CDIM;

    if (q >= NK) {
        // zero pad patches
        for (int idx = threadIdx.x; idx < 64 * 64; idx += 256) {
            const int tok = idx >> 6, c4 = idx & 63;
            const int i = tok >> 3, j = tok & 7;
            const size_t drow = (size_t)(kh * PH + i) * WNEW + kw * PH + j;
            *(float4*)(out + outBase + drow * CDIM + c4 * 4) = make_float4(0.f, 0.f, 0.f, 0.f);
        }
        return;
    }

    const int p  = top_idx[b * NK + q];
    const int py = p >> 4;
    const int px = p & 15;
    const size_t inBase = (size_t)b * (HIMG * WIMG) * CDIM;
    for (int idx = threadIdx.x; idx < 64 * 64; idx += 256) {
        const int tok = idx >> 6, c4 = idx & 63;
        const int i = tok >> 3, j = tok & 7;
        const size_t srow = (size_t)(py * PH + i) * WIMG + px * PH + j;
        const size_t drow = (size_t)(kh * PH + i) * WNEW + kw * PH + j;

        const int idx2 = idx + 256;
        if (idx2 < 64 * 64) {
            const int tok2 = idx2 >> 6;
            const int i2 = tok2 >> 3, j2 = tok2 & 7;
            const size_t srow2 = (size_t)(py * PH + i2) * WIMG + px * PH + j2;
            __builtin_prefetch(tokens + inBase + srow2 * CDIM + (idx2 & 63) * 4, 0, 3);
        }

        *(float4*)(out + outBase + drow * CDIM + c4 * 4) =
            *(const float4*)(tokens + inBase + srow * CDIM + c4 * 4);
    }
}

// ---------------------------------------------------------------------------
extern "C" void kernel_launch(void* const* d_in, const int* in_sizes, int n_in,
                              void* d_out, int out_size, void* d_ws, size_t ws_size,
                              hipStream_t stream)
{
    (void)in_sizes; (void)n_in; (void)out_size; (void)ws_size;
    const float* tokens = (const float*)d_in[0];
    const float* W1     = (const float*)d_in[1];
    const float* b1     = (const float*)d_in[2];
    const float* W2     = (const float*)d_in[3];
    const float* b2     = (const float*)d_in[4];

    float* scores  = (float*)d_ws;                                          // NB*NP floats
    int*   top_idx = (int*)((char*)d_ws + (size_t)NB * NP * sizeof(float)); // NB*NK ints
    float* out     = (float*)d_out;

    score_kernel <<<NB * NP,        128, 0, stream>>>(tokens, W1, b1, W2, b2, scores);
    topk_kernel  <<<NB,             256, 0, stream>>>(scores, top_idx);
    gather_kernel<<<NB * NKH * NKW, 256, 0, stream>>>(tokens, top_idx, out);
}